// GCN_7430293422974
// MI455X (gfx1250) — compile-verified
//
#include <hip/hip_runtime.h>

#define N_NODES_C 100000
#define N_EDGES_C 600000
#define D_FEAT_C  128

typedef float v4f __attribute__((ext_vector_type(4)));
typedef int   v2i __attribute__((ext_vector_type(2)));

#define AS1 __attribute__((address_space(1)))
#define AS3 __attribute__((address_space(3)))

#if __has_builtin(__builtin_amdgcn_global_load_async_to_lds_b64) && \
    __has_builtin(__builtin_amdgcn_s_wait_asynccnt)
#define GCN_USE_ASYNC 1
#else
#define GCN_USE_ASYNC 0
#endif

// ---------------------------------------------------------------------------
// Kernel 1: zero degree histogram + the global segment cursor.
// (No zeroing of `out` needed: the gather kernel fully writes every row.)
// ---------------------------------------------------------------------------
__global__ void __launch_bounds__(256) gcn_zero_init(unsigned* __restrict__ deg,
                                                     unsigned* __restrict__ counter, int n) {
  const int i = blockIdx.x * blockDim.x + threadIdx.x;
  const int stride = gridDim.x * blockDim.x;
  for (int k = i; k < n; k += stride) deg[k] = 0u;
  if (i == 0) *counter = 0u;
}

// ---------------------------------------------------------------------------
// Kernel 2: degree histogram over destination nodes.
// ---------------------------------------------------------------------------
__global__ void __launch_bounds__(256) gcn_degree(const long long* __restrict__ edst,
                                                  unsigned* __restrict__ deg, int e) {
  const int i = blockIdx.x * blockDim.x + threadIdx.x;
  const int stride = gridDim.x * blockDim.x;
  for (int k = i; k < e; k += stride) {
    atomicAdd(&deg[(int)edst[k]], 1u);
  }
}

// ---------------------------------------------------------------------------
// Kernel 3: segment allocation via wave32 prefix scan + one atomic per wave,
// fused with deg^{-1/2}. Produces start[] (stable for gather) and cursor[]
// (consumed by the CSR fill). Precise 1/sqrt for numerics (one-time pass).
// ---------------------------------------------------------------------------
__global__ void __launch_bounds__(256) gcn_alloc(const unsigned* __restrict__ deg,
                                                 unsigned* __restrict__ start,
                                                 unsigned* __restrict__ cursor,
                                                 float* __restrict__ dinv,
                                                 unsigned* __restrict__ counter, int n) {
  const int i = blockIdx.x * 256 + threadIdx.x;
  const int lane = threadIdx.x & 31;

  unsigned d = (i < n) ? deg[i] : 0u;

  // Inclusive wave scan (wave32).
  unsigned incl = d;
#pragma unroll
  for (int off = 1; off < 32; off <<= 1) {
    unsigned t = __shfl_up(incl, off, 32);
    if (lane >= off) incl += t;
  }
  const unsigned total = __shfl(incl, 31, 32);

  unsigned base = 0u;
  if (lane == 31) base = atomicAdd(counter, total);
  base = __shfl(base, 31, 32);

  if (i < n) {
    const unsigned s0 = base + incl - d;  // exclusive position
    start[i] = s0;
    cursor[i] = s0;
    dinv[i] = d ? (1.0f / sqrtf((float)d)) : 0.f;
  }
}

// ---------------------------------------------------------------------------
// Kernel 4: CSR fill. Edge (src,dst) int64 pairs are staged into LDS via
// CDNA5 async global->LDS copies (ASYNCcnt / s_wait_asynccnt), then each
// thread claims a slot in its destination's segment with one u32 atomic.
// ---------------------------------------------------------------------------
__global__ void __launch_bounds__(256) gcn_csr_fill(const long long* __restrict__ esrc,
                                                    const long long* __restrict__ edst,
                                                    unsigned* __restrict__ cursor,
                                                    int* __restrict__ csr_src) {
  __shared__ long long lsrc[256];
  __shared__ long long ldst[256];
  const int tid = threadIdx.x;
  const int ntiles = (N_EDGES_C + 255) >> 8;

  for (int tile = blockIdx.x; tile < ntiles; tile += gridDim.x) {
    const int eid = (tile << 8) + tid;

#if GCN_USE_ASYNC
    if (eid < N_EDGES_C) {
      __builtin_amdgcn_global_load_async_to_lds_b64(
          (AS1 v2i*)&esrc[eid], (AS3 v2i*)&lsrc[tid], 0, 0);
      __builtin_amdgcn_global_load_async_to_lds_b64(
          (AS1 v2i*)&edst[eid], (AS3 v2i*)&ldst[tid], 0, 0);
    }
    __builtin_amdgcn_s_wait_asynccnt(0);
    __syncthreads();
#else
    if (eid < N_EDGES_C) {
      lsrc[tid] = esrc[eid];
      ldst[tid] = edst[eid];
    }
    __syncthreads();
#endif

    if (eid < N_EDGES_C) {
      const int s = (int)lsrc[tid];
      const int d = (int)ldst[tid];
      const unsigned pos = atomicAdd(&cursor[d], 1u);
      csr_src[pos] = s;
    }
    __syncthreads();
  }
}

// ---------------------------------------------------------------------------
// Kernel 5: atomic-free gather + fused ReLU. One wave per destination node.
// Lanes load up to 32 neighbor ids + dinv gathers in parallel, prefetch the
// source rows, then broadcast with v_readlane (uniform index -> no LDS pipe)
// and accumulate the 128-float row as float4 per lane. Single
// global_store_b128 per output row.
// ---------------------------------------------------------------------------
__global__ void __launch_bounds__(256) gcn_gather_relu(const float* __restrict__ x,
                                                       const int* __restrict__ csr_src,
                                                       const unsigned* __restrict__ start,
                                                       const unsigned* __restrict__ deg,
                                                       const float* __restrict__ dinv,
                                                       float* __restrict__ out, int n) {
  const int lane = threadIdx.x & 31;
  const int wave = threadIdx.x >> 5;
  const int i = blockIdx.x * 8 + wave;
  if (i >= n) return;

  const unsigned b = start[i];
  const int len = (int)deg[i];
  const float dv = dinv[i];

  v4f acc = {0.f, 0.f, 0.f, 0.f};

  for (int cb = 0; cb < len; cb += 32) {
    const int c = min(32, len - cb);  // wave-uniform
    int s_l = 0;
    float n_l = 0.f;
    if (lane < c) {
      s_l = csr_src[b + cb + lane];
      n_l = dv * dinv[s_l];  // parallel dinv gather
      // Pull all c source rows toward cache before the broadcast loop.
      __builtin_prefetch(x + (long)s_l * D_FEAT_C, 0, 0);
    }
    for (int k = 0; k < c; ++k) {  // k is wave-uniform -> readlane is legal
      const int sk = __builtin_amdgcn_readlane(s_l, k);
      const float nk =
          __int_as_float(__builtin_amdgcn_readlane(__float_as_int(n_l), k));
      const v4f row = ((const v4f*)(x + (long)sk * D_FEAT_C))[lane];
      acc += row * nk;
    }
  }

  acc.x = fmaxf(acc.x, 0.f);
  acc.y = fmaxf(acc.y, 0.f);
  acc.z = fmaxf(acc.z, 0.f);
  acc.w = fmaxf(acc.w, 0.f);
  ((v4f*)(out + (long)i * D_FEAT_C))[lane] = acc;
}

extern "C" void kernel_launch(void* const* d_in, const int* in_sizes, int n_in,
                              void* d_out, int out_size, void* d_ws, size_t ws_size,
                              hipStream_t stream) {
  (void)in_sizes; (void)n_in; (void)out_size; (void)ws_size;

  const float* x = (const float*)d_in[0];
  const long long* eidx = (const long long*)d_in[1];  // int64 [2, E]
  const long long* esrc = eidx;                        // edge_index[0] (source j)
  const long long* edst = eidx + N_EDGES_C;            // edge_index[1] (target i)
  float* out = (float*)d_out;

  // Workspace layout (~4.0 MB):
  unsigned* deg = (unsigned*)d_ws;                 // [N] u32
  unsigned* start = deg + N_NODES_C;               // [N] u32
  unsigned* cursor = start + N_NODES_C;            // [N] u32
  float* dinv = (float*)(cursor + N_NODES_C);      // [N] f32
  unsigned* counter = (unsigned*)(dinv + N_NODES_C);  // [1] u32 (padded)
  int* csr_src = (int*)(counter + 64);             // [E] i32

  gcn_zero_init<<<256, 256, 0, stream>>>(deg, counter, N_NODES_C);
  gcn_degree<<<1024, 256, 0, stream>>>(edst, deg, N_EDGES_C);

  const int alloc_blocks = (N_NODES_C + 255) / 256;  // 391
  gcn_alloc<<<alloc_blocks, 256, 0, stream>>>(deg, start, cursor, dinv, counter, N_NODES_C);

  const int ntiles = (N_EDGES_C + 255) / 256;  // 2344
  gcn_csr_fill<<<ntiles, 256, 0, stream>>>(esrc, edst, cursor, csr_src);

  const int gather_blocks = (N_NODES_C + 7) / 8;  // 12500
  gcn_gather_relu<<<gather_blocks, 256, 0, stream>>>(x, csr_src, start, deg, dinv, out, N_NODES_C);
}